// RDGCNN_35407710388862
// MI455X (gfx1250) — compile-verified
//
#include <hip/hip_runtime.h>
#include <math.h>

// ---------------------------------------------------------------------------
// RDGCNN for MI455X (gfx1250). f32 WMMA 16x16x4 used for the kNN Gram matrix,
// the edge-conv, and the final projection. kNN selection via per-wave LDS
// argmin over a 16x2048 distance strip held in CDNA5's 320KB LDS.
// Two-pass BN statistics (recompute instead of 335MB y-scratch).
// cin is a template parameter (9 / 32) so all fragment-load guards and K-loop
// bounds are compile-time: no EXEC-juggling in the WMMA inner loops.
// ---------------------------------------------------------------------------

#define BN    8
#define NPTS  2048
#define KNN   40
#define CHO   32
#define EPSV  1e-5f
#define SLOPE 0.2f
#define CNTF  ((float)(BN * NPTS * KNN))   // 655360

typedef float v2f __attribute__((ext_vector_type(2)));
typedef float v8f __attribute__((ext_vector_type(8)));

__device__ __forceinline__ v8f wmma4(v2f a, v2f b, v8f c) {
  // D = A(16x4,f32) * B(4x16,f32) + C(16x16,f32)   -> v_wmma_f32_16x16x4_f32
  return __builtin_amdgcn_wmma_f32_16x16x4_f32(
      /*neg_a=*/false, a, /*neg_b=*/false, b,
      /*c_mod=*/(short)0, c, /*reuse_a=*/false, /*reuse_b=*/false);
}

// ---------------------------------------------------------------------------
// Kernel 0: zero the BN statistic accumulators (4 stages x [sum32|sumsq32]).
// ---------------------------------------------------------------------------
__global__ void zero_stats_kernel(float* __restrict__ s) {
  if (threadIdx.x < 256) s[threadIdx.x] = 0.0f;
}

// ---------------------------------------------------------------------------
// Kernel 1: kNN indices via WMMA Gram matrix.
// grid = B * (N/16) blocks of 256 threads (8 waves).
// Phase A: cooperative xx[m] = sum_c x[c,m]^2 into LDS.
// Phase B: each wave computes 16 column tiles of the 16-row distance strip
//          with v_wmma_f32_16x16x4_f32 (A-fragments hoisted, K-loop fully
//          unrolled):  dist[n][m] = xx[n] - 2 * inner(n,m) + xx[m]
//          strip kept in 128KB LDS (2 WGs fit in CDNA5's 320KB WGP LDS).
// Phase C: each wave runs 40 rounds of masked wave-parallel argmin on 2 rows.
// ---------------------------------------------------------------------------
template <int CIN>
__global__ void knn_kernel(const float* __restrict__ x, int* __restrict__ idx) {
  constexpr int CPAD = (CIN + 3) & ~3;
  constexpr int NCH  = CPAD / 4;
  __shared__ float dist[16][NPTS];   // 128 KB distance strip
  __shared__ float xx[NPTS];         // 8 KB squared norms
  const int b  = blockIdx.x >> 7;            // N/16 = 128 row tiles per batch
  const int n0 = (blockIdx.x & 127) * 16;
  const float* xb = x + (size_t)b * CIN * NPTS;
  const int tid = threadIdx.x;

  // Phase A: squared norms
  for (int m = tid; m < NPTS; m += 256) {
    float s = 0.0f;
    __builtin_prefetch(&xb[m + 256], 0, 0);
#pragma unroll
    for (int c = 0; c < CIN; ++c) {
      float v = xb[c * NPTS + m];
      s = fmaf(v, v, s);
    }
    xx[m] = s;
  }
  __syncthreads();

  const int wave = tid >> 5;
  const int lane = tid & 31;
  const int half = lane >> 4;   // 0: K rows cc,cc+1 ; 1: K rows cc+2,cc+3
  const int l15  = lane & 15;

  // Hoisted A-fragments: rows = n0..n0+15, one fragment per K chunk.
  v2f afrag[NCH];
#pragma unroll
  for (int q = 0; q < NCH; ++q) {
    const int cc = q * 4 + (half << 1);
    afrag[q].x = (cc     < CIN) ? xb[cc * NPTS       + n0 + l15] : 0.0f;
    afrag[q].y = (cc + 1 < CIN) ? xb[(cc + 1) * NPTS + n0 + l15] : 0.0f;
  }

  // Phase B: inner products via WMMA, convert to distances
  for (int ct = wave; ct < NPTS / 16; ct += 8) {
    const int m0 = ct * 16;
    v8f acc = {};
#pragma unroll
    for (int q = 0; q < NCH; ++q) {
      const int cc = q * 4 + (half << 1);
      v2f bf;
      bf.x = (cc     < CIN) ? xb[cc * NPTS       + m0 + l15] : 0.0f;
      bf.y = (cc + 1 < CIN) ? xb[(cc + 1) * NPTS + m0 + l15] : 0.0f;
      acc  = wmma4(afrag[q], bf, acc);
    }
    const int col  = m0 + l15;
    const float xm = xx[col];
#pragma unroll
    for (int v = 0; v < 8; ++v) {
      const int r = half * 8 + v;                // D layout: M = half*8+v
      dist[r][col] = xx[n0 + r] - 2.0f * acc[v] + xm;
    }
  }
  __syncthreads();

  // Phase C: top-40 smallest per row via repeated argmin (2 rows per wave)
#pragma unroll
  for (int rr = 0; rr < 2; ++rr) {
    const int r = wave * 2 + rr;
    const int n = n0 + r;
    float* drow = dist[r];
    int* orow = idx + ((size_t)b * NPTS + n) * KNN;
    for (int kk = 0; kk < KNN; ++kk) {
      float best = 3.4e38f;
      int   bi   = NPTS;
      for (int m = lane; m < NPTS; m += 32) {
        float v = drow[m];
        if (v < best || (v == best && m < bi)) { best = v; bi = m; }
      }
#pragma unroll
      for (int off = 16; off > 0; off >>= 1) {
        float ov = __shfl_xor(best, off, 32);
        int   oi = __shfl_xor(bi,   off, 32);
        if (ov < best || (ov == best && oi < bi)) { best = ov; bi = oi; }
      }
      if (lane == 0) orow[kk] = bi;
      drow[bi] = 3.4e38f;   // all lanes agree on bi; mask it out
    }
  }
}

// ---------------------------------------------------------------------------
// Edge-conv core: y = W(32 x 2cin) * e(2cin x 48cols) for one point, via
// f32 WMMA 16x16x4.  Column tiles t=0..2 hold neighbors k = t*16+lane15
// (cols >= 40 padded with self, masked later).  M tiles m=0..1 (channels).
//   PASS2=false: accumulate per-channel sum / sumsq into stats (atomics).
//   PASS2=true : normalize, leaky-relu, max over k, add residual, store.
// grid = B*N/4 blocks of 128 threads (one point per wave).
// ---------------------------------------------------------------------------
template <int CIN>
__device__ __forceinline__ float edge_val(const float* __restrict__ xb,
                                          int r, int j, int n) {
  if (r >= 2 * CIN) return 0.0f;                 // K-padding rows
  if (r < CIN) return xb[r * NPTS + j] - xb[r * NPTS + n];   // feat - center
  return xb[(r - CIN) * NPTS + n];                           // center
}

template <int CIN, bool PASS2>
__global__ void edge_pass_kernel(const float* __restrict__ x,
                                 const float* __restrict__ W,
                                 const float* __restrict__ gamma,
                                 const float* __restrict__ beta,
                                 float* __restrict__ stats,
                                 const int* __restrict__ idx,
                                 float* __restrict__ out,
                                 const float* __restrict__ resid) {
  constexpr int C2   = 2 * CIN;
  constexpr int WPAD = (C2 + 3) & ~3;
  constexpr int NCH  = WPAD / 4;
  __shared__ float Wlds[CHO * WPAD];
  __shared__ float bsum[CHO];
  __shared__ float bsq[CHO];

  for (int i = threadIdx.x; i < CHO * WPAD; i += blockDim.x) {
    int o = i / WPAD, c = i - o * WPAD;
    Wlds[i] = (c < C2) ? W[o * C2 + c] : 0.0f;
  }
  if (!PASS2 && threadIdx.x < CHO) { bsum[threadIdx.x] = 0.0f; bsq[threadIdx.x] = 0.0f; }
  __syncthreads();

  const int wave = threadIdx.x >> 5;
  const int lane = threadIdx.x & 31;
  const int half = lane >> 4;        // 0: K rows c,c+1 ; 1: K rows c+2,c+3
  const int l15  = lane & 15;
  const int row  = blockIdx.x * 4 + wave;
  const int b    = row / NPTS;
  const int n    = row % NPTS;
  const float* xb   = x + (size_t)b * CIN * NPTS;
  const int*   jrow = idx + (size_t)row * KNN;

  int jt[3];
#pragma unroll
  for (int t = 0; t < 3; ++t) {
    int col = t * 16 + l15;
    jt[t] = (col < KNN) ? jrow[col] : n;
  }

  v8f acc[3][2];
#pragma unroll
  for (int t = 0; t < 3; ++t) { acc[t][0] = {}; acc[t][1] = {}; }

#pragma unroll
  for (int q = 0; q < NCH; ++q) {
    const int cc = q * 4 + (half << 1);
    v2f a0, a1;
    a0.x = Wlds[l15 * WPAD + cc];
    a0.y = Wlds[l15 * WPAD + cc + 1];
    a1.x = Wlds[(16 + l15) * WPAD + cc];
    a1.y = Wlds[(16 + l15) * WPAD + cc + 1];
#pragma unroll
    for (int t = 0; t < 3; ++t) {
      v2f bf;
      bf.x = edge_val<CIN>(xb, cc,     jt[t], n);
      bf.y = edge_val<CIN>(xb, cc + 1, jt[t], n);
      acc[t][0] = wmma4(a0, bf, acc[t][0]);
      acc[t][1] = wmma4(a1, bf, acc[t][1]);
    }
  }

  if (!PASS2) {
    // --- statistics pass: per-channel sum and sum-of-squares over valid k ---
#pragma unroll
    for (int m = 0; m < 2; ++m) {
#pragma unroll
      for (int v = 0; v < 8; ++v) {
        float s = 0.0f, q = 0.0f;
#pragma unroll
        for (int t = 0; t < 3; ++t) {
          int col = t * 16 + l15;
          float val = acc[t][m][v];
          if (col < KNN) { s += val; q += val * val; }
        }
#pragma unroll
        for (int off = 1; off < 16; off <<= 1) {   // reduce within 16-lane half
          s += __shfl_xor(s, off, 32);
          q += __shfl_xor(q, off, 32);
        }
        if (l15 == 0) {
          int ch = m * 16 + half * 8 + v;
          atomicAdd(&bsum[ch], s);
          atomicAdd(&bsq[ch],  q);
        }
      }
    }
    __syncthreads();
    if (threadIdx.x < CHO) {
      atomicAdd(&stats[threadIdx.x],       bsum[threadIdx.x]);
      atomicAdd(&stats[CHO + threadIdx.x], bsq[threadIdx.x]);
    }
  } else {
    // --- output pass: batchnorm + leaky-relu + max over k (+ residual) ---
#pragma unroll
    for (int m = 0; m < 2; ++m) {
#pragma unroll
      for (int v = 0; v < 8; ++v) {
        int ch = m * 16 + half * 8 + v;
        float mean = stats[ch] * (1.0f / CNTF);
        float var  = stats[CHO + ch] * (1.0f / CNTF) - mean * mean;
        float rs   = rsqrtf(var + EPSV);
        float sc   = gamma[ch] * rs;
        float sh   = beta[ch] - mean * sc;
        float vm   = -3.4e38f;
#pragma unroll
        for (int t = 0; t < 3; ++t) {
          int col = t * 16 + l15;
          float val = acc[t][m][v] * sc + sh;
          val = (val >= 0.0f) ? val : SLOPE * val;
          if (col < KNN) vm = fmaxf(vm, val);
        }
#pragma unroll
        for (int off = 1; off < 16; off <<= 1)
          vm = fmaxf(vm, __shfl_xor(vm, off, 32));
        if (l15 == 0) {
          size_t o = (size_t)b * CHO * NPTS + (size_t)ch * NPTS + n;
          float r = resid ? resid[o] : 0.0f;
          out[o] = vm + r;
        }
      }
    }
  }
}

// ---------------------------------------------------------------------------
// Kernel 3: final projection out = W5(128x128) * cat([x1;x2;x3;x4]) per (b,n).
// grid = B * (N/16) blocks of 256 threads; wave w = output channel tile w.
// ---------------------------------------------------------------------------
__device__ __forceinline__ float cat_val(const float* __restrict__ x1,
                                         const float* __restrict__ x2,
                                         const float* __restrict__ x3,
                                         const float* __restrict__ x4,
                                         int b, int c, int n) {
  const float* xp = (c < 32) ? x1 : (c < 64) ? x2 : (c < 96) ? x3 : x4;
  return xp[(size_t)b * CHO * NPTS + (size_t)(c & 31) * NPTS + n];
}

__global__ void final_proj_kernel(const float* __restrict__ W5,
                                  const float* __restrict__ x1,
                                  const float* __restrict__ x2,
                                  const float* __restrict__ x3,
                                  const float* __restrict__ x4,
                                  float* __restrict__ out) {
  const int b     = blockIdx.x >> 7;       // N/16 = 128 tiles per batch
  const int ntile = blockIdx.x & 127;
  const int wave  = threadIdx.x >> 5;      // M tile 0..7
  const int lane  = threadIdx.x & 31;
  const int half  = lane >> 4;
  const int l15   = lane & 15;
  const int n     = ntile * 16 + l15;

  v8f acc = {};
#pragma unroll
  for (int c0 = 0; c0 < 128; c0 += 4) {
    const int cc = c0 + (half << 1);
    v2f a, bf;
    const int mrow = wave * 16 + l15;
    a.x  = W5[mrow * 128 + cc];
    a.y  = W5[mrow * 128 + cc + 1];
    bf.x = cat_val(x1, x2, x3, x4, b, cc,     n);
    bf.y = cat_val(x1, x2, x3, x4, b, cc + 1, n);
    acc  = wmma4(a, bf, acc);
  }
#pragma unroll
  for (int v = 0; v < 8; ++v) {
    int ch = wave * 16 + half * 8 + v;
    out[(size_t)b * 128 * NPTS + (size_t)ch * NPTS + n] = acc[v];
  }
}

// ---------------------------------------------------------------------------
// Host launch: stage pipeline x -> x1 -> x2 -> x3 -> x4 -> final projection.
// ---------------------------------------------------------------------------
extern "C" void kernel_launch(void* const* d_in, const int* in_sizes, int n_in,
                              void* d_out, int out_size, void* d_ws, size_t ws_size,
                              hipStream_t stream) {
  const float* x  = (const float*)d_in[0];
  const float* Wm[4] = {(const float*)d_in[1], (const float*)d_in[2],
                        (const float*)d_in[3], (const float*)d_in[4]};
  const float* W5 = (const float*)d_in[5];
  // dict order after W5: g1,b1,g2,b2,g3,b3,g4,b4
  const float* g[4] = {(const float*)d_in[6], (const float*)d_in[8],
                       (const float*)d_in[10], (const float*)d_in[12]};
  const float* be[4] = {(const float*)d_in[7], (const float*)d_in[9],
                        (const float*)d_in[11], (const float*)d_in[13]};

  char* ws = (char*)d_ws;
  float* stats = (float*)ws;                               // 4*64 floats
  int*   idxb  = (int*)(ws + 1024);                        // B*N*K ints
  size_t off   = 1024 + (size_t)BN * NPTS * KNN * sizeof(int);
  float* xb[4];
  for (int s = 0; s < 4; ++s)
    xb[s] = (float*)(ws + off + (size_t)s * BN * CHO * NPTS * sizeof(float));

  const int nb  = BN * NPTS / 4;        // one point per wave, 4 waves per block
  const int nkb = BN * (NPTS / 16);     // one 16-row strip per block
  zero_stats_kernel<<<1, 256, 0, stream>>>(stats);

  // Stage 1 (cin = 9)
  knn_kernel<9><<<nkb, 256, 0, stream>>>(x, idxb);
  edge_pass_kernel<9, false><<<nb, 128, 0, stream>>>(
      x, Wm[0], g[0], be[0], stats + 0 * 64, idxb, nullptr, nullptr);
  edge_pass_kernel<9, true><<<nb, 128, 0, stream>>>(
      x, Wm[0], g[0], be[0], stats + 0 * 64, idxb, xb[0], nullptr);

  // Stages 2..4 (cin = 32, residual)
  for (int s = 1; s < 4; ++s) {
    knn_kernel<32><<<nkb, 256, 0, stream>>>(xb[s - 1], idxb);
    edge_pass_kernel<32, false><<<nb, 128, 0, stream>>>(
        xb[s - 1], Wm[s], g[s], be[s], stats + s * 64, idxb, nullptr, nullptr);
    edge_pass_kernel<32, true><<<nb, 128, 0, stream>>>(
        xb[s - 1], Wm[s], g[s], be[s], stats + s * 64, idxb, xb[s], xb[s - 1]);
  }

  final_proj_kernel<<<BN * (NPTS / 16), 256, 0, stream>>>(
      W5, xb[0], xb[1], xb[2], xb[3], (float*)d_out);
}